// Bayesian_Network_54142357733520
// MI455X (gfx1250) — compile-verified
//
#include <hip/hip_runtime.h>

// Bayesian GCN (2-layer, T=4 masked samples) for MI455X / gfx1250.
// Dominant cost: 5 edge aggregations (L2-resident gather + f32 scatter atomics).
// GEMMs are tiny (7.4 GFLOP total) -> keep fp32 precision, V_WMMA_F32_16X16X4_F32.

#define IN_DIM  128
#define HID_DIM 128
#define OUT_DIM 16
#define SK      132   // padded k-stride (dwords), SK % 64 == 4 -> conflict-free b64 B loads

typedef __attribute__((ext_vector_type(2))) float v2f;
typedef __attribute__((ext_vector_type(8))) float v8f;

// ---------------------------------------------------------------------------
// Edge aggregation: out[dst] += x[src] * w   (x,out are [N,128] fp32)
// One wave per edge; each lane owns 4 consecutive floats (32*4 = 128).
// Edge metadata is wave-uniform -> readfirstlane => scalar loads.
// Gather: global_load_b128; scatter: 4x global_atomic_add_f32 (no return).
// ---------------------------------------------------------------------------
__global__ __launch_bounds__(256)
void agg_kernel(const float* __restrict__ x, const int* __restrict__ src,
                const int* __restrict__ dst, const float* __restrict__ w,
                float* __restrict__ out, int E) {
    long long tid = (long long)blockIdx.x * blockDim.x + threadIdx.x;
    int e    = __builtin_amdgcn_readfirstlane((int)(tid >> 5));  // wave-uniform edge id
    int lane = (int)(tid & 31);
    if (e >= E) return;
    int   s  = src[e];
    int   d  = dst[e];
    float wt = w[e];
    const float4 v = *(const float4*)(x + (size_t)s * IN_DIM + lane * 4);
    float* o = out + (size_t)d * IN_DIM + lane * 4;
    atomicAdd(o + 0, v.x * wt);
    atomicAdd(o + 1, v.y * wt);
    atomicAdd(o + 2, v.z * wt);
    atomicAdd(o + 3, v.w * wt);
}

// ---------------------------------------------------------------------------
// h = relu(h0 @ (layer1 .* mask))   [N,128] x [128,128]
// Block = 128 threads (4 waves). Masked weights staged TRANSPOSED in LDS:
//   wT[n*SK + k] = mask[k,n] ? layer1[k,n] : 0
// fp32 16x16x4 fragment layouts (ISA 7.12.2):
//   A(16x4): lane = m + 16*half holds A[m][k+2h], A[m][k+2h+1]  (one b64)
//   B(4x16): lane = n + 16*half holds B[k+2h][n], B[k+2h+1][n]
//            = wT[n*SK + k+2h], wT[n*SK + k+2h+1]               (one b64)
//   C/D:     vgpr r, lane = n + 16*half -> element (M = r + 8*half, N = n)
// SK % 64 == 4 => lane (n,half) bank-pair = {4n+2h, 4n+2h+1}: all 64 banks,
// zero conflicts on the ds_load_b64s.
// ---------------------------------------------------------------------------
__global__ __launch_bounds__(128)
void gemm1_relu_kernel(const float* __restrict__ h0,
                       const float* __restrict__ layer1,
                       const unsigned char* __restrict__ mask,
                       float* __restrict__ h, int n) {
    __shared__ float wT[HID_DIM * SK];                 // 67.6 KB of 320 KB LDS
    for (int i = threadIdx.x; i < IN_DIM * HID_DIM; i += 128) {
        int k = i >> 7, c = i & 127;                   // layer1[k][c], row-major
        wT[c * SK + k] = mask[i] ? layer1[i] : 0.0f;
    }
    __syncthreads();

    int wave = threadIdx.x >> 5;
    int lane = threadIdx.x & 31;
    int rowbase = (blockIdx.x * 4 + wave) * 16;
    if (rowbase >= n) return;                          // wave-uniform: EXEC all-1s at WMMA

    int half = lane >> 4;
    int m    = lane & 15;                              // A row / B column within tile
    const float* arow = h0 + (size_t)(rowbase + m) * IN_DIM;

    v8f acc[8];
    const v8f zero = {0.f, 0.f, 0.f, 0.f, 0.f, 0.f, 0.f, 0.f};
#pragma unroll
    for (int nt = 0; nt < 8; ++nt) acc[nt] = zero;

    for (int k = 0; k < IN_DIM; k += 4) {
        v2f a = *(const v2f*)(arow + k + 2 * half);    // global b64, 8B aligned
#pragma unroll
        for (int nt = 0; nt < 8; ++nt) {
            v2f b = *(const v2f*)(&wT[(nt * 16 + m) * SK + k + 2 * half]);  // ds b64
            acc[nt] = __builtin_amdgcn_wmma_f32_16x16x4_f32(
                false, a, false, b, (short)0, acc[nt], false, false);
        }
    }

#pragma unroll
    for (int nt = 0; nt < 8; ++nt)
#pragma unroll
        for (int r = 0; r < 8; ++r) {
            float vv = acc[nt][r];
            h[(size_t)(rowbase + r + 8 * half) * HID_DIM + nt * 16 + m] =
                vv > 0.f ? vv : 0.f;
        }
}

// ---------------------------------------------------------------------------
// out = o @ (layer2 .* mask)   [N,128] x [128,16] -> [N,16]
// Same scheme, single 16-wide output tile per wave; weights transposed in LDS.
// ---------------------------------------------------------------------------
__global__ __launch_bounds__(128)
void gemm2_kernel(const float* __restrict__ o,
                  const float* __restrict__ layer2,
                  const unsigned char* __restrict__ mask,
                  float* __restrict__ out, int n) {
    __shared__ float wT[OUT_DIM * SK];                 // 8.4 KB
    for (int i = threadIdx.x; i < HID_DIM * OUT_DIM; i += 128) {
        int k = i >> 4, c = i & 15;                    // layer2[k][c], row-major
        wT[c * SK + k] = mask[i] ? layer2[i] : 0.0f;
    }
    __syncthreads();

    int wave = threadIdx.x >> 5;
    int lane = threadIdx.x & 31;
    int rowbase = (blockIdx.x * 4 + wave) * 16;
    if (rowbase >= n) return;

    int half = lane >> 4;
    int m    = lane & 15;
    const float* arow = o + (size_t)(rowbase + m) * HID_DIM;

    v8f acc = {0.f, 0.f, 0.f, 0.f, 0.f, 0.f, 0.f, 0.f};
    for (int k = 0; k < HID_DIM; k += 4) {
        v2f a = *(const v2f*)(arow + k + 2 * half);
        v2f b = *(const v2f*)(&wT[m * SK + k + 2 * half]);
        acc = __builtin_amdgcn_wmma_f32_16x16x4_f32(
            false, a, false, b, (short)0, acc, false, false);
    }
#pragma unroll
    for (int r = 0; r < 8; ++r)
        out[(size_t)(rowbase + r + 8 * half) * OUT_DIM + m] = acc[r];
}

// ---------------------------------------------------------------------------
// Orchestration. Inputs (setup_inputs order):
//  0 feature[N,128] f32 | 1 edge_weight[E] f32 | 2 layer1[128,128] f32
//  3 layer2[128,16] f32 | 4 src[E] i32 | 5 dst[E] i32
//  6 mask1[T,128,128] bool(u8) | 7 mask2[T,128,16] bool(u8)
// Output: [T,N,16] f32. Workspace: h0 | h_t | o_t (3 x N*128 f32 = 76.8 MB).
// ---------------------------------------------------------------------------
extern "C" void kernel_launch(void* const* d_in, const int* in_sizes, int n_in,
                              void* d_out, int out_size, void* d_ws, size_t ws_size,
                              hipStream_t stream) {
    const float* feature        = (const float*)d_in[0];
    const float* ew             = (const float*)d_in[1];
    const float* layer1         = (const float*)d_in[2];
    const float* layer2         = (const float*)d_in[3];
    const int*   src            = (const int*)d_in[4];
    const int*   dst            = (const int*)d_in[5];
    const unsigned char* mask1  = (const unsigned char*)d_in[6];
    const unsigned char* mask2  = (const unsigned char*)d_in[7];
    float* out = (float*)d_out;

    const int N = in_sizes[0] / IN_DIM;
    const int E = in_sizes[1];
    const int T = in_sizes[6] / (IN_DIM * HID_DIM);

    float* h0 = (float*)d_ws;
    float* h  = h0 + (size_t)N * HID_DIM;
    float* o  = h  + (size_t)N * HID_DIM;
    const size_t matBytes = (size_t)N * HID_DIM * sizeof(float);

    const unsigned aggBlocks  = (unsigned)(((long long)E * 32 + 255) / 256);
    const int      gemmBlocks = (N + 63) / 64;   // 4 waves x 16 rows per block

    hipMemsetAsync(h0, 0, matBytes, stream);
    agg_kernel<<<aggBlocks, 256, 0, stream>>>(feature, src, dst, ew, h0, E);

    for (int t = 0; t < T; ++t) {
        gemm1_relu_kernel<<<gemmBlocks, 128, 0, stream>>>(
            h0, layer1, mask1 + (size_t)t * IN_DIM * HID_DIM, h, N);
        hipMemsetAsync(o, 0, matBytes, stream);
        agg_kernel<<<aggBlocks, 256, 0, stream>>>(h, src, dst, ew, o, E);
        gemm2_kernel<<<gemmBlocks, 128, 0, stream>>>(
            o, layer2, mask2 + (size_t)t * HID_DIM * OUT_DIM,
            out + (size_t)t * N * OUT_DIM, N);
    }
}